// MultiheadAttentionStream_18794776888018
// MI455X (gfx1250) — compile-verified
//
#include <hip/hip_runtime.h>
#include <stdint.h>

// ---------------------------------------------------------------------------
// Streaming attention (alpha-decay push step), MI455X / gfx1250.
// Memory-bound (AI ~= 0.5 FLOP/B): k+v = 537 MB -> ~23 us floor @ 23.3 TB/s.
// Design: partitioned batched GEMV + frozen-max softmax, NT loads, async
// LDS staging (CDNA5 async path), WMMA f32 16x16x4 for the push dot.
// Workspace use: ~4.27 MB of d_ws (dots, partial max/Z/QKV).
// ---------------------------------------------------------------------------

#define BH 128
#define S  8192
#define D  64
#define P  8             // partitions of S per batch-head
#define SP (S / P)       // 1024 keys per partition
#define NT1 256          // threads for k1/k2
#define ALPHA 0.999f

typedef float f2  __attribute__((ext_vector_type(2)));
typedef float f4  __attribute__((ext_vector_type(4)));
typedef float v2f __attribute__((ext_vector_type(2)));
typedef float v8f __attribute__((ext_vector_type(8)));

// workspace layout (in floats)
#define WS_DOTS 0
#define WS_MAX  (BH * S)
#define WS_Z    (WS_MAX + BH * P)
#define WS_QKV  (WS_Z + BH * P)
// total floats = BH*S + 2*BH*P + BH*P*D = 1,116,160  (~4.27 MB)

#if __has_builtin(__builtin_amdgcn_global_load_async_to_lds_b128)
#define HAVE_ASYNC 1
// Exact parameter types per the round-1 diagnostic:
//   param0: __attribute__((vector_size(16))) int  __attribute__((address_space(1)))*
//   param1: (by symmetry) same vector type in address_space(3)
typedef int i4v __attribute__((vector_size(16)));
typedef __attribute__((address_space(1))) i4v g_i4v;
typedef __attribute__((address_space(3))) i4v l_i4v;
#endif

__device__ __forceinline__ void wait_async0() {
#if defined(HAVE_ASYNC)
#if __has_builtin(__builtin_amdgcn_s_wait_asynccnt)
  __builtin_amdgcn_s_wait_asynccnt(0);
#else
  asm volatile("s_wait_asynccnt 0x0" ::: "memory");
#endif
#endif
}

// ---------------------------------------------------------------------------
// K1: dots[bh,s] = q[bh,:] . k[bh,s,:]  +  per-partition max
// grid (P, BH), 256 threads, 4 keys/thread
// ---------------------------------------------------------------------------
__global__ __launch_bounds__(NT1) void k1_dots(const float* __restrict__ q,
                                               const float* __restrict__ k,
                                               float* __restrict__ ws) {
  const int p = blockIdx.x, bh = blockIdx.y, t = threadIdx.x;
  const int sp0 = p * SP;
  __shared__ float red[NT1];

  const f4* qv = (const f4*)(q + bh * D);
  f4 q4[16];
#pragma unroll
  for (int i = 0; i < 16; ++i) q4[i] = qv[i];

  float* dots = ws + WS_DOTS + (size_t)bh * S;
  float tmax = -3.402823466e38f;
#pragma unroll
  for (int j = 0; j < SP / NT1; ++j) {
    const int s = sp0 + t + j * NT1;
    if (j + 1 < SP / NT1)
      __builtin_prefetch(k + ((size_t)bh * S + s + NT1) * D, 0, 0);
    const f4* kr = (const f4*)(k + ((size_t)bh * S + s) * D);
    float d = 0.f;
#pragma unroll
    for (int i = 0; i < 16; ++i) {
      const f4 kv = __builtin_nontemporal_load(kr + i);  // streamed once: NT
      d += q4[i].x * kv.x; d += q4[i].y * kv.y;
      d += q4[i].z * kv.z; d += q4[i].w * kv.w;
    }
    dots[s] = d;
    tmax = fmaxf(tmax, d);
  }
  red[t] = tmax;
  __syncthreads();
  for (int o = NT1 / 2; o > 0; o >>= 1) {
    if (t < o) red[t] = fmaxf(red[t], red[t + o]);
    __syncthreads();
  }
  if (t == 0) ws[WS_MAX + bh * P + p] = red[0];
}

// ---------------------------------------------------------------------------
// K2: e = exp(dot - m + mask); partial Z; partial QKV = sum_s e_s * v[s,:]
// grid (P, BH), 256 threads = 8 waves. Mask partition staged to LDS via the
// CDNA5 async-to-LDS path (overlapped with the max fetch; waited on before the dot reads).
// ---------------------------------------------------------------------------
__global__ __launch_bounds__(NT1) void k2_expv(const float* __restrict__ v,
                                               const float* __restrict__ mask,
                                               float* __restrict__ ws) {
  const int p = blockIdx.x, bh = blockIdx.y, t = threadIdx.x;
  const int sp0 = p * SP;
  __shared__ float lmask[SP];   // 4 KB
  __shared__ float le[SP];      // 4 KB
  __shared__ float red[NT1];    // 1 KB
  __shared__ float part[8][D];  // 2 KB

  const float* mrow = mask + (size_t)bh * S + sp0;
#if defined(HAVE_ASYNC)
  // 256 lanes x 16B = the whole 4 KB mask partition in one async instruction
  __builtin_amdgcn_global_load_async_to_lds_b128(
      (g_i4v*)(mrow + 4 * t), (l_i4v*)(&lmask[4 * t]), 0, 0);
#else
  ((f4*)lmask)[t] = ((const f4*)mrow)[t];
#endif

  // global max over the P partition maxima (tiny, overlaps the async copy)
  float m = -3.402823466e38f;
#pragma unroll
  for (int i = 0; i < P; ++i) m = fmaxf(m, ws[WS_MAX + bh * P + i]);

  wait_async0();
  __syncthreads();

  const float* dots = ws + WS_DOTS + (size_t)bh * S;
  float zp = 0.f;
#pragma unroll
  for (int j = 0; j < SP / NT1; ++j) {
    const int sl = t + j * NT1;
    const float e = __expf(dots[sp0 + sl] - m + lmask[sl]);
    le[sl] = e;
    zp += e;
  }
  red[t] = zp;
  __syncthreads();
  for (int o = NT1 / 2; o > 0; o >>= 1) {
    if (t < o) red[t] += red[t + o];
    __syncthreads();
  }
  if (t == 0) ws[WS_Z + bh * P + p] = red[0];

  // partial QKV: wave w owns 128 keys; lane l owns columns {2l, 2l+1}.
  // Each wave iteration reads one contiguous 256 B v-row (perfect coalescing).
  const int w = t >> 5, l = t & 31;
  const int keys = SP / 8;  // 128
  f2 acc; acc.x = 0.f; acc.y = 0.f;
#pragma unroll 4
  for (int jj = 0; jj < keys; ++jj) {
    const int sl = w * keys + jj;
    const float e = le[sl];  // wave-uniform LDS broadcast
    const f2 vv = __builtin_nontemporal_load(
        (const f2*)(v + ((size_t)bh * S + sp0 + sl) * D) + l);
    acc.x += e * vv.x;
    acc.y += e * vv.y;
  }
  part[w][2 * l]     = acc.x;
  part[w][2 * l + 1] = acc.y;
  __syncthreads();
  if (t < D) {
    float s = 0.f;
#pragma unroll
    for (int i = 0; i < 8; ++i) s += part[i][t];
    ws[WS_QKV + ((size_t)bh * P + p) * D + t] = s;
  }
}

// ---------------------------------------------------------------------------
// K3: combine partials + push step. The push dot q_push.k_push (K=64) runs on
// the matrix pipe: 16 chained v_wmma_f32_16x16x4_f32 with A/B replicated per
// the documented f32 16x4 / 4x16 lane layouts (wave 0, EXEC all-ones).
// grid (BH), 64 threads (thread t = output column d).
// ---------------------------------------------------------------------------
__global__ __launch_bounds__(64) void k3_push(const float* __restrict__ q_push,
                                              const float* __restrict__ k_push,
                                              const float* __restrict__ v_push,
                                              const float* __restrict__ ws,
                                              float* __restrict__ out) {
  const int bh = blockIdx.x, t = threadIdx.x;
  __shared__ float sdot;

  float m = -3.402823466e38f, Z = 0.f, qkv = 0.f;
#pragma unroll
  for (int i = 0; i < P; ++i) {
    m = fmaxf(m, ws[WS_MAX + bh * P + i]);
    Z += ws[WS_Z + bh * P + i];
    qkv += ws[WS_QKV + ((size_t)bh * P + i) * D + t];
  }

  const float* qp = q_push + bh * D;
  const float* kp = k_push + bh * D;
#if __has_builtin(__builtin_amdgcn_wmma_f32_16x16x4_f32)
  if (t < 32) {  // wave 0 only; uniform branch -> EXEC all 1s inside
    // A 16x4 f32: lane<16 holds (K0,K1) of row M=lane; lane>=16 holds (K2,K3).
    // B 4x16 f32: same split per lane-half across N. Replicating q over M and
    // k over N makes every D[m][n] equal dot(q,k) over this K-slab.
    const int half = (t >> 4) & 1;
    v8f c = {0.f, 0.f, 0.f, 0.f, 0.f, 0.f, 0.f, 0.f};
#pragma unroll
    for (int it = 0; it < 16; ++it) {
      const int kb = 4 * it + 2 * half;
      v2f a, b;
      a.x = qp[kb]; a.y = qp[kb + 1];
      b.x = kp[kb]; b.y = kp[kb + 1];
      c = __builtin_amdgcn_wmma_f32_16x16x4_f32(false, a, false, b, (short)0,
                                                c, false, false);
    }
    if (t == 0) sdot = c[0];
  }
#else
  if (t == 0) {
    float d = 0.f;
    for (int i = 0; i < D; ++i) d += qp[i] * kp[i];
    sdot = d;
  }
#endif
  __syncthreads();

  const float ep = __expf(sdot - m);  // frozen running max, per reference
  out[bh * D + t] =
      (ALPHA * qkv + ep * v_push[bh * D + t]) / (ALPHA * Z + ep);
}

// ---------------------------------------------------------------------------
extern "C" void kernel_launch(void* const* d_in, const int* in_sizes, int n_in,
                              void* d_out, int out_size, void* d_ws,
                              size_t ws_size, hipStream_t stream) {
  const float* q      = (const float*)d_in[0];
  const float* k      = (const float*)d_in[1];
  const float* v      = (const float*)d_in[2];
  const float* q_push = (const float*)d_in[3];
  const float* k_push = (const float*)d_in[4];
  const float* v_push = (const float*)d_in[5];
  const float* mask   = (const float*)d_in[6];
  float* ws  = (float*)d_ws;
  float* out = (float*)d_out;
  (void)in_sizes; (void)n_in; (void)out_size; (void)ws_size;

  dim3 grid(P, BH, 1);
  k1_dots<<<grid, NT1, 0, stream>>>(q, k, ws);
  k2_expv<<<grid, NT1, 0, stream>>>(v, mask, ws);
  k3_push<<<BH, 64, 0, stream>>>(q_push, k_push, v_push, ws, out);
}